// ShardedCSRSparseRetrievalModel_48928267436213
// MI455X (gfx1250) — compile-verified
//
#include <hip/hip_runtime.h>

// ---------------- problem constants (from reference) ----------------
#define NUM_DOCS     1000000
#define VOCAB        30522
#define VOCAB_PAD    30528          // multiple of 4 for float4 zero-fill
#define NNZ_PER_ROW  64
#define Q_NNZ        32
#define SHARDS       4
#define TOP_K        100
#define ROWS_PER_SHARD (NUM_DOCS / SHARDS)

#define HIST_BINS    65536          // top-16 bits of order-preserving float key
#define CAP          16384          // candidate cap per shard (generous)

// workspace layout (bytes)
#define SCORES_OFF   0u                                  // 1M f32  = 4,000,000
#define HIST_OFF     4000000u                            // 4*65536 u32 = 1,048,576
#define CNT_OFF      (HIST_OFF + HIST_BINS * SHARDS * 4) // 4 u32 (zeroed with hist)
#define CANDS_OFF    (CNT_OFF + 64u)                     // 4*CAP f32
#define CANDI_OFF    (CANDS_OFF + SHARDS * CAP * 4u)     // 4*CAP i32
#define STS_OFF      (CANDI_OFF + SHARDS * CAP * 4u)     // 400 f32
#define STI_OFF      (STS_OFF + SHARDS * TOP_K * 4u)     // 400 i32

typedef float v2f __attribute__((ext_vector_type(2)));
typedef float v8f __attribute__((ext_vector_type(8)));

// ------------- order-preserving (score,index) packing -------------
__device__ __forceinline__ unsigned ord_f32(float f) {
    unsigned u = __float_as_uint(f);
    return (u & 0x80000000u) ? ~u : (u | 0x80000000u);   // monotone float order
}
__device__ __forceinline__ unsigned long long pack_key(float f, unsigned idx) {
    return ((unsigned long long)ord_f32(f) << 32) |
           (unsigned long long)(0xFFFFFFFFu - idx);       // tie -> lower idx wins
}
__device__ __forceinline__ float key_score(unsigned long long key) {
    unsigned u = (unsigned)(key >> 32);
    unsigned bits = (u & 0x80000000u) ? (u ^ 0x80000000u) : ~u;
    return __uint_as_float(bits);
}
__device__ __forceinline__ unsigned key_index(unsigned long long key) {
    return 0xFFFFFFFFu - (unsigned)(key & 0xFFFFFFFFu);
}

// ---------------------------------------------------------------
// Kernel 1: dense-query-in-LDS SpMV with WMMA row-sum reduction.
// One wave = 16 rows. Lane l: row = base + (l&15), K-group g = l>>4.
// Bucket->nnz partition is free, so lane group g owns the contiguous
// nnz span [32g, 32g+32): acc0 = first 16, acc1 = last 16. All doc
// traffic becomes 16B-aligned global_load_b128; one
// v_wmma_f32_16x16x4_f32 against all-ones B then produces the 16
// full-precision row sums (32-bit A-matrix 16x4 VGPR layout).
// ---------------------------------------------------------------
__global__ void spmv_scores_kernel(const float* __restrict__ dv,
                                   const int*   __restrict__ di,
                                   const float* __restrict__ qv,
                                   const int*   __restrict__ qi,
                                   float*       __restrict__ scores) {
    extern __shared__ float sq[];          // VOCAB_PAD floats = 122,112 B
    const int tid = threadIdx.x;

    // zero the dense query table (b128 LDS stores)
    float4* sq4 = reinterpret_cast<float4*>(sq);
    for (int i = tid; i < VOCAB_PAD / 4; i += blockDim.x) {
        sq4[i] = make_float4(0.f, 0.f, 0.f, 0.f);
    }
    __syncthreads();

    // deterministic sparse->dense scatter (fixed order for duplicates)
    if (tid == 0) {
        for (int t = 0; t < Q_NNZ; ++t) sq[qi[t]] += qv[t];
    }
    __syncthreads();

    const int lane = tid & 31;
    const int wave = tid >> 5;
    const int globalWave  = blockIdx.x * (blockDim.x >> 5) + wave;
    const int waveRowBase = globalWave * 16;
    if (waveRowBase >= NUM_DOCS) return;   // wave-uniform tail guard (after barriers)

    const int m = lane & 15;
    const int g = lane >> 4;

    const long long rowOff = (long long)(waveRowBase + m) * NNZ_PER_ROW + 32 * g;
    const float4* vp = reinterpret_cast<const float4*>(dv + rowOff);
    const int4*   ip = reinterpret_cast<const int4*>(di + rowOff);

    float acc0 = 0.f, acc1 = 0.f;
#pragma unroll
    for (int t = 0; t < 4; ++t) {          // nnz [32g, 32g+16) -> K-bucket 2g
        float4 v = vp[t];                  // global_load_b128
        int4   c = ip[t];                  // global_load_b128
        acc0 = fmaf(v.x, sq[c.x], acc0);   // ds_load_b32 gathers
        acc0 = fmaf(v.y, sq[c.y], acc0);
        acc0 = fmaf(v.z, sq[c.z], acc0);
        acc0 = fmaf(v.w, sq[c.w], acc0);
    }
#pragma unroll
    for (int t = 4; t < 8; ++t) {          // nnz [32g+16, 32g+32) -> K-bucket 2g+1
        float4 v = vp[t];
        int4   c = ip[t];
        acc1 = fmaf(v.x, sq[c.x], acc1);
        acc1 = fmaf(v.y, sq[c.y], acc1);
        acc1 = fmaf(v.z, sq[c.z], acc1);
        acc1 = fmaf(v.w, sq[c.w], acc1);
    }

    // WMMA reduction: D[m,n] = sum_k A[m,k] * 1 + 0  (all n identical)
    v2f a; a[0] = acc0; a[1] = acc1;
    v2f b; b[0] = 1.0f; b[1] = 1.0f;
    v8f c8 = {};
    v8f d = __builtin_amdgcn_wmma_f32_16x16x4_f32(
        false, a, false, b, (short)0, c8, false, false);

    // D layout: VGPR v, lanes 0-15 -> M=v ; lanes 16-31 -> M=v+8.
    const int k = lane & 7;
    float s = (k >= 4) ? ((k >= 6) ? (k == 7 ? d[7] : d[6])
                                   : (k == 5 ? d[5] : d[4]))
                       : ((k >= 2) ? (k == 3 ? d[3] : d[2])
                                   : (k == 1 ? d[1] : d[0]));
    if ((lane & 8) == 0) {
        scores[waveRowBase + k + (g ? 8 : 0)] = s;
    }
}

// ---------------------------------------------------------------
// Kernel 2: per-shard histogram over top-16 bits of the ordered key.
// ~93% of scores are exactly 0 (query hits are rare) -> skip zeros:
// only ~65K atomics total, and the selection threshold is > 0 anyway.
// ---------------------------------------------------------------
__global__ void hist_kernel(const float* __restrict__ scores,
                            unsigned*    __restrict__ hist) {
    const int i = blockIdx.x * blockDim.x + threadIdx.x;
    if (i >= NUM_DOCS) return;
    const float f = scores[i];
    if (f != 0.0f) {
        const int shard = i / ROWS_PER_SHARD;
        atomicAdd(&hist[shard * HIST_BINS + (ord_f32(f) >> 16)], 1u);
    }
}

// ---------------------------------------------------------------
// Kernel 3: per shard, find threshold bin B (cumulative-from-top first
// reaches TOP_K), then collect all candidates with prefix >= B.
// ---------------------------------------------------------------
__global__ void shard_select_kernel(const float*    __restrict__ scores,
                                    const unsigned* __restrict__ hist,
                                    unsigned*       __restrict__ candCnt,
                                    float*          __restrict__ candS,
                                    int*            __restrict__ candI) {
    __shared__ unsigned chunkSum[512];
    __shared__ unsigned sB;
    const int shard = blockIdx.x;
    const int tid   = threadIdx.x;
    const unsigned* h = hist + shard * HIST_BINS;
    const int CHUNK = HIST_BINS / 512;                 // 128 bins per thread

    unsigned sum = 0;
    for (int b = 0; b < CHUNK; ++b) sum += h[tid * CHUNK + b];
    chunkSum[tid] = sum;
    __syncthreads();

    if (tid == 0) {
        unsigned cum = 0, B = 0;
        for (int c = 511; c >= 0; --c) {
            if (cum + chunkSum[c] < TOP_K) { cum += chunkSum[c]; continue; }
            for (int b = c * CHUNK + CHUNK - 1; b >= c * CHUNK; --b) {
                cum += h[b];
                if (cum >= TOP_K) { B = (unsigned)b; break; }
            }
            break;
        }
        sB = B;   // B==0 (collect all nonzero) if shard has < TOP_K nonzeros
    }
    __syncthreads();

    const unsigned B = sB;
    const int rbase = shard * ROWS_PER_SHARD;
    for (int i = tid; i < ROWS_PER_SHARD; i += blockDim.x) {
        const float f = scores[rbase + i];
        if (f != 0.0f && (ord_f32(f) >> 16) >= B) {
            const unsigned pos = atomicAdd(&candCnt[shard], 1u);
            if (pos < CAP) {
                candS[shard * CAP + pos] = f;
                candI[shard * CAP + pos] = rbase + i;
            }
        }
    }
}

// ---------------------------------------------------------------
// Kernel 4: exact per-shard top-100 over the candidates.
// One wave per shard; barrier-free shfl_xor max-reduction; knockout.
// Selection keys on (score desc, doc-index asc) -> order-independent
// of the nondeterministic atomic-append candidate order.
// ---------------------------------------------------------------
__global__ void shard_top_kernel(float* __restrict__ candS,
                                 int*   __restrict__ candI,
                                 const unsigned* __restrict__ candCnt,
                                 float* __restrict__ topS,
                                 int*   __restrict__ topI) {
    const int shard = blockIdx.x;
    const int lane  = threadIdx.x;
    int n = (int)candCnt[shard];
    if (n > CAP) n = CAP;
    float* cs = candS + shard * CAP;
    int*   ci = candI + shard * CAP;

    for (int k = 0; k < TOP_K; ++k) {
        unsigned long long best = 0ull;
        int bslot = -1;
        for (int i = lane; i < n; i += 32) {
            const unsigned long long key = pack_key(cs[i], (unsigned)ci[i]);
            if (key > best) { best = key; bslot = i; }
        }
        for (int off = 16; off > 0; off >>= 1) {
            const unsigned long long ok = __shfl_xor(best, off, 32);
            const int os = __shfl_xor(bslot, off, 32);
            if (ok > best) { best = ok; bslot = os; }
        }
        if (lane == 0) {
            if (best != 0ull) {
                topS[shard * TOP_K + k] = key_score(best);
                topI[shard * TOP_K + k] = (int)key_index(best);
            } else {                                // unreachable in practice
                topS[shard * TOP_K + k] = -1e30f;
                topI[shard * TOP_K + k] = shard * ROWS_PER_SHARD;
            }
            if (bslot >= 0) cs[bslot] = -1.0f;      // knockout (same-wave st->ld ordered)
        }
    }
}

// ---------------------------------------------------------------
// Kernel 5: merge 4x100 -> global top-100 (one wave, all in LDS).
// Tie-break = position in the concatenated shard lists (reference
// merge semantics). d_out = [100 scores f32][100 indices as f32].
// ---------------------------------------------------------------
__global__ void merge_kernel(const float* __restrict__ topS,
                             const int*   __restrict__ topI,
                             float*       __restrict__ out) {
    __shared__ float ls[SHARDS * TOP_K];
    __shared__ int   li[SHARDS * TOP_K];
    const int lane = threadIdx.x;
    for (int i = lane; i < SHARDS * TOP_K; i += 32) {
        ls[i] = topS[i];
        li[i] = topI[i];
    }
    __syncthreads();
    for (int k = 0; k < TOP_K; ++k) {
        unsigned long long best = 0ull;
        int bslot = 0;
        for (int i = lane; i < SHARDS * TOP_K; i += 32) {
            const unsigned long long key = pack_key(ls[i], (unsigned)i);
            if (key > best) { best = key; bslot = i; }
        }
        for (int off = 16; off > 0; off >>= 1) {
            const unsigned long long ok = __shfl_xor(best, off, 32);
            const int os = __shfl_xor(bslot, off, 32);
            if (ok > best) { best = ok; bslot = os; }
        }
        if (lane == 0) {
            out[k]         = ls[bslot];
            out[TOP_K + k] = (float)li[bslot];
            ls[bslot] = -1e30f;                     // knockout
        }
        __syncthreads();                            // single wave: ordering fence
    }
}

// ---------------------------------------------------------------
extern "C" void kernel_launch(void* const* d_in, const int* in_sizes, int n_in,
                              void* d_out, int out_size, void* d_ws, size_t ws_size,
                              hipStream_t stream) {
    const float* doc_values  = (const float*)d_in[0];
    const float* q_values    = (const float*)d_in[1];
    const int*   doc_indices = (const int*)  d_in[2];
    // d_in[3] = row_ids: implied by nnz position (j / 64), unused
    const int*   q_indices   = (const int*)  d_in[4];

    char* ws = (char*)d_ws;
    float*    scores  = (float*)   (ws + SCORES_OFF);
    unsigned* hist    = (unsigned*)(ws + HIST_OFF);
    unsigned* candCnt = (unsigned*)(ws + CNT_OFF);
    float*    candS   = (float*)   (ws + CANDS_OFF);
    int*      candI   = (int*)     (ws + CANDI_OFF);
    float*    topS    = (float*)   (ws + STS_OFF);
    int*      topI    = (int*)     (ws + STI_OFF);

    // zero histogram + candidate counters (stream-ordered, capture-safe)
    hipMemsetAsync(ws + HIST_OFF, 0, HIST_BINS * SHARDS * 4 + 64, stream);

    const int smemBytes = VOCAB_PAD * (int)sizeof(float);   // 122,112 B of LDS
    hipFuncSetAttribute((const void*)spmv_scores_kernel,
                        hipFuncAttributeMaxDynamicSharedMemorySize, smemBytes);

    // 256 threads = 8 waves = 128 rows per block; ceil(62500 waves / 8) blocks
    const int nblocks = (NUM_DOCS / 16 + 7) / 8;            // 7813
    spmv_scores_kernel<<<nblocks, 256, smemBytes, stream>>>(
        doc_values, doc_indices, q_values, q_indices, scores);

    hist_kernel<<<(NUM_DOCS + 255) / 256, 256, 0, stream>>>(scores, hist);

    shard_select_kernel<<<SHARDS, 512, 0, stream>>>(scores, hist, candCnt, candS, candI);

    shard_top_kernel<<<SHARDS, 32, 0, stream>>>(candS, candI, candCnt, topS, topI);

    merge_kernel<<<1, 32, 0, stream>>>(topS, topI, (float*)d_out);
}